// LSTM_71236327571486
// MI455X (gfx1250) — compile-verified
//
#include <hip/hip_runtime.h>
#include <hip/hip_bf16.h>

typedef __attribute__((ext_vector_type(2))) float v2f;
typedef __attribute__((ext_vector_type(8))) float v8f;

#define INPUT_SZ 20
#define HID 14
#define G4 56     // 4*HID (live gate columns)
#define XGS 64    // padded row stride of xg (cols 56..63 are garbage)

// ---------- helpers: raw cross-lane ops (wave32) ----------
__device__ __forceinline__ int f2i(float f) { return __builtin_bit_cast(int, f); }
__device__ __forceinline__ float i2f(int i) { return __builtin_bit_cast(float, i); }

// broadcast value held in lane `srcLane` to all lanes (uniform, v_readlane)
__device__ __forceinline__ float lane_bcast(float v, int srcLane) {
  return i2f(__builtin_amdgcn_readlane(f2i(v), srcLane));
}
// per-lane gather: this lane reads `v` from lane index (wraps mod 32)
__device__ __forceinline__ float lane_gather(float v, int srcLane) {
  return i2f(__builtin_amdgcn_ds_bpermute(srcLane << 2, f2i(v)));
}

// =====================================================================
// Phase 1: xg[t][n] = sum_k x[t][k]*W_ih[n][k] + (b_ih[n]+b_hh[n])
// WMMA f32 16x16x4: per wave, one 16(time) x 64(gate,padded) tile,
// 5 K-chunks x 4 N-tiles = 20 v_wmma_f32_16x16x4_f32 per tile.
// Bias is folded into the accumulator init; output rows padded to XGS.
// =====================================================================
__global__ __launch_bounds__(256) void lstm_xgemm(
    const float* __restrict__ x, const float* __restrict__ Wih,
    const float* __restrict__ bih, const float* __restrict__ bhh,
    float* __restrict__ xg, int T) {
  const int lane = threadIdx.x & 31;
  const int wave = threadIdx.x >> 5;
  const int wavesPerBlock = blockDim.x >> 5;
  const int m = lane & 15;            // M row within tile (both lane halves)
  const int khalf = (lane >> 4) << 1; // 0 for lanes 0-15, 2 for lanes 16-31
  const int rbase = (lane >> 4) << 3; // C VGPR r -> row r + rbase

  // Preload B (W_ih transposed): B[nt][ks] covers K = ks*4+khalf..+1, N = (lane&15)+nt*16
  v2f B[4][5];
  float bias[4];
  for (int nt = 0; nt < 4; ++nt) {
    int n = (lane & 15) + nt * 16;
    bias[nt] = (n < G4) ? (bih[n] + bhh[n]) : 0.f;
    for (int ks = 0; ks < 5; ++ks) {
      int k = ks * 4 + khalf;
      float b0 = 0.f, b1 = 0.f;
      if (n < G4) { b0 = Wih[n * INPUT_SZ + k]; b1 = Wih[n * INPUT_SZ + k + 1]; }
      v2f b = {b0, b1};
      B[nt][ks] = b;
    }
  }

  const int numTiles = (T + 15) >> 4;
  for (int tile = blockIdx.x * wavesPerBlock + wave; tile < numTiles;
       tile += gridDim.x * wavesPerBlock) {
    const int t0 = tile << 4;
    int rows = T - t0; if (rows > 16) rows = 16;
    const float* xr = x + (long)(t0 + m) * INPUT_SZ + khalf;

    v2f A[5];
    if (rows == 16) {
      // full tile: branch-free vector loads (8B aligned: even offsets)
#pragma unroll
      for (int ks = 0; ks < 5; ++ks)
        A[ks] = *(const v2f*)(xr + ks * 4);
    } else {
      const bool mvalid = (m < rows);
#pragma unroll
      for (int ks = 0; ks < 5; ++ks) {
        float a0 = mvalid ? xr[ks * 4] : 0.f;
        float a1 = mvalid ? xr[ks * 4 + 1] : 0.f;
        v2f a = {a0, a1};
        A[ks] = a;
      }
    }

#pragma unroll
    for (int nt = 0; nt < 4; ++nt) {
      // accumulator starts at the bias (same value for every row of this column)
      float bv = bias[nt];
      v8f c = {bv, bv, bv, bv, bv, bv, bv, bv};
#pragma unroll
      for (int ks = 0; ks < 5; ++ks) {
        c = __builtin_amdgcn_wmma_f32_16x16x4_f32(
            false, A[ks], false, B[nt][ks], (short)0, c, false, false);
      }
      // C layout: VGPR r -> M = r + rbase, N = lane&15 (+16*nt)
      const int col = (lane & 15) + nt * 16;
      float* dst = xg + (long)(t0 + rbase) * XGS + col;
      if (rows == 16) {
#pragma unroll
        for (int r = 0; r < 8; ++r)
          dst[(long)r * XGS] = c[r];           // padded: no column guard
      } else {
#pragma unroll
        for (int r = 0; r < 8; ++r)
          if (r + rbase < rows) dst[(long)r * XGS] = c[r];
      }
    }
  }
}

// =====================================================================
// Phase 2: single-wave sequential LSTM scan over `len` steps.
// lane j owns gate rows j and j+32 (valid while j+32 < 56).
// gates order: i=g[0:14], f=g[14:28], gg=g[28:42], o=g[42:56]
// =====================================================================
__global__ __launch_bounds__(32, 1) void lstm_scan(
    const float* __restrict__ xg, const float* __restrict__ Whh,
    const float* __restrict__ h0, const float* __restrict__ c0,
    const float* __restrict__ fcw, const float* __restrict__ fcb,
    float* __restrict__ state, float* __restrict__ out,
    int first, int len, int last) {
  const int j = threadIdx.x;           // 0..31
  const float L2E = 1.44269504088896340736f;

  // Preload recurrent weights for this lane's two gate rows
  float Wlo[HID], Whi[HID];
#pragma unroll
  for (int k = 0; k < HID; ++k) {
    Wlo[k] = Whh[j * HID + k];                         // rows 0..31 all valid
    Whi[k] = (j < G4 - 32) ? Whh[(j + 32) * HID + k] : 0.f;
  }
  // Unified activation: act = A/(1+exp2(nB*x)) + C  (sigmoid or tanh)
  const bool lo_tanh = (j >= 28);            // lo gate index j: [28,32) is gg
  const bool hi_tanh = (j < 10);             // hi gate index j+32: [32,42) is gg
  const float Alo = lo_tanh ? 2.f : 1.f, Clo = lo_tanh ? -1.f : 0.f;
  const float Ahi = hi_tanh ? 2.f : 1.f, Chi = hi_tanh ? -1.f : 0.f;
  const float nBlo = (lo_tanh ? -2.f : -1.f) * L2E;
  const float nBhi = (hi_tanh ? -2.f : -1.f) * L2E;

  float h, c;
  if (first) {
    h = (j < HID) ? h0[j] : 0.f;
    c = (j < HID) ? c0[j] : 0.f;
  } else {
    h = state[j];
    c = state[32 + j];
  }

  const float* p = xg;
  for (int t = 0; t < len; ++t, p += XGS) {
    __builtin_prefetch(p + 2 * XGS, 0, 0);         // stream xg ahead (global_prefetch_b8)
    float glo0 = p[j];                             // gate row j pre-activation
    float ghi0 = (j < G4 - 32) ? p[32 + j] : 0.f;  // gate row j+32
    float glo1 = 0.f, ghi1 = 0.f;
    // g += W_hh @ h : two independent 7-deep FMA chains (readlanes independent)
#pragma unroll
    for (int k = 0; k < HID; k += 2) {
      float hk0 = lane_bcast(h, k);
      float hk1 = lane_bcast(h, k + 1);
      glo0 = fmaf(hk0, Wlo[k], glo0);
      glo1 = fmaf(hk1, Wlo[k + 1], glo1);
      ghi0 = fmaf(hk0, Whi[k], ghi0);
      ghi1 = fmaf(hk1, Whi[k + 1], ghi1);
    }
    float glo = glo0 + glo1;
    float ghi = ghi0 + ghi1;
    float alo = Alo / (1.f + __builtin_amdgcn_exp2f(nBlo * glo)) + Clo;
    float ahi = Ahi / (1.f + __builtin_amdgcn_exp2f(nBhi * ghi)) + Chi;
    // recombine into lanes 0..13 (k = j):
    // i[k]  = alo (local);  f[k] = lane 14+k lo;
    // gg[k] = k<4 ? lane 28+k lo : lane k-4 hi;  o[k] = lane 10+k hi
    float f_k = lane_gather(alo, j + 14);
    float tga = lane_gather(alo, j + 28);
    float tgb = lane_gather(ahi, (j + 28) & 31);   // (j-4) mod 32
    float tg  = (j < 4) ? tga : tgb;
    float o_k = lane_gather(ahi, j + 10);
    float cn = fmaf(f_k, c, alo * tg);
    float tc = 2.f / (1.f + __builtin_amdgcn_exp2f(-2.f * L2E * cn)) - 1.f;
    h = o_k * tc;
    c = cn;
  }
  state[j] = h;
  state[32 + j] = c;

  if (last) {  // out = h . fc_w + fc_b  (butterfly reduction over wave32)
    float pr = (j < HID) ? h * fcw[j] : 0.f;
#pragma unroll
    for (int off = 16; off >= 1; off >>= 1) pr += lane_gather(pr, j ^ off);
    if (j == 0) out[0] = pr + fcb[0];
  }
}

// =====================================================================
extern "C" void kernel_launch(void* const* d_in, const int* in_sizes, int n_in,
                              void* d_out, int out_size, void* d_ws, size_t ws_size,
                              hipStream_t stream) {
  const float* x   = (const float*)d_in[0];
  const float* Wih = (const float*)d_in[1];
  const float* Whh = (const float*)d_in[2];
  const float* bih = (const float*)d_in[3];
  const float* bhh = (const float*)d_in[4];
  const float* h0  = (const float*)d_in[5];
  const float* c0  = (const float*)d_in[6];
  const float* fcw = (const float*)d_in[7];
  const float* fcb = (const float*)d_in[8];
  (void)n_in; (void)out_size;

  const int T = in_sizes[0] / INPUT_SZ;  // 500000

  // workspace layout: [0..63] state floats (h:0..13, c:32..45), then xg chunk
  float* state = (float*)d_ws;
  float* xg = state + 64;
  long maxChunk = 0;
  if (ws_size > 64 * sizeof(float))
    maxChunk = (long)((ws_size - 64 * sizeof(float)) / (XGS * sizeof(float)));
  long chunk = maxChunk;
  if (chunk > T) chunk = T;
  chunk &= ~15L;            // multiple of 16 for GEMM tiling
  if (chunk < 16) chunk = 16;

  for (long s0 = 0; s0 < T; s0 += chunk) {
    int len = (int)(((T - s0) < chunk) ? (T - s0) : chunk);
    int numTiles = (len + 15) / 16;
    int blocks = (numTiles + 7) / 8;   // 8 waves per block, 1 tile per wave
    if (blocks > 2048) blocks = 2048;
    if (blocks < 1) blocks = 1;
    lstm_xgemm<<<blocks, 256, 0, stream>>>(x + s0 * INPUT_SZ, Wih, bih, bhh, xg, len);
    lstm_scan<<<1, 32, 0, stream>>>(xg, Whh, h0, c0, fcw, fcb, state,
                                    (float*)d_out,
                                    (s0 == 0) ? 1 : 0, len,
                                    ((s0 + len) >= T) ? 1 : 0);
  }
}